// ViewTransformerLSSBEVDepth_28544352649880
// MI455X (gfx1250) — compile-verified
//
#include <hip/hip_runtime.h>
#include <hip/hip_bf16.h>

// ---------------------------------------------------------------------------
// Problem constants (from the reference)
// ---------------------------------------------------------------------------
#define BATCH   4
#define NCAM    6
#define BN      (BATCH * NCAM)     // 24
#define CIN     512
#define FH      16
#define FW      44
#define FHW     (FH * FW)          // 704
#define DEPTHC  59
#define CT      64
#define OC      (DEPTHC + CT)      // 123
#define OCPAD   128
#define NX0     128
#define NX1     128
#define MTILE   32                 // pixels per block (2 WMMA M-tiles)
#define BEV_ELEMS   ((size_t)BATCH * CT * NX0 * NX1)   // 4,194,304
#define DEPTH_ELEMS ((size_t)BN * DEPTHC * FHW)        // 996,864

typedef __attribute__((ext_vector_type(2))) float v2f;
typedef __attribute__((ext_vector_type(8))) float v8f;

// ---------------------------------------------------------------------------
// small 3x3 helpers
// ---------------------------------------------------------------------------
__device__ __forceinline__ void inv3(const float* a, float* o) {
    float c00 = a[4]*a[8] - a[5]*a[7];
    float c01 = a[5]*a[6] - a[3]*a[8];
    float c02 = a[3]*a[7] - a[4]*a[6];
    float det = a[0]*c00 + a[1]*c01 + a[2]*c02;
    float id  = 1.0f / det;
    o[0] = c00 * id;
    o[1] = (a[2]*a[7] - a[1]*a[8]) * id;
    o[2] = (a[1]*a[5] - a[2]*a[4]) * id;
    o[3] = c01 * id;
    o[4] = (a[0]*a[8] - a[2]*a[6]) * id;
    o[5] = (a[2]*a[3] - a[0]*a[5]) * id;
    o[6] = c02 * id;
    o[7] = (a[1]*a[6] - a[0]*a[7]) * id;
    o[8] = (a[0]*a[4] - a[1]*a[3]) * id;
}
__device__ __forceinline__ void mul3(const float* a, const float* b, float* o) {
    for (int i = 0; i < 3; ++i)
        for (int j = 0; j < 3; ++j)
            o[i*3+j] = a[i*3+0]*b[0*3+j] + a[i*3+1]*b[1*3+j] + a[i*3+2]*b[2*3+j];
}

// ---------------------------------------------------------------------------
// Kernel 1: per-camera matrix precompute.
// mats[bn*21 + 0..8]  = inv(post_rots)
// mats[bn*21 + 9..17] = bda @ rots @ inv(intrins)
// mats[bn*21 +18..20] = bda @ trans
// ---------------------------------------------------------------------------
__global__ void setup_mats_kernel(const float* __restrict__ rots,
                                  const float* __restrict__ trans,
                                  const float* __restrict__ intrins,
                                  const float* __restrict__ post_rots,
                                  const float* __restrict__ bda,
                                  float* __restrict__ mats) {
    int bn = blockIdx.x * blockDim.x + threadIdx.x;
    if (bn >= BN) return;
    int b = bn / NCAM;
    float R[9], K[9], P[9], A[9];
    for (int i = 0; i < 9; ++i) {
        R[i] = rots[bn*9 + i];
        K[i] = intrins[bn*9 + i];
        P[i] = post_rots[bn*9 + i];
        A[i] = bda[b*9 + i];
    }
    float Kinv[9], Pinv[9], T1[9], comb[9];
    inv3(K, Kinv);
    inv3(P, Pinv);
    mul3(R, Kinv, T1);
    mul3(A, T1, comb);
    float tv[3];
    for (int i = 0; i < 3; ++i)
        tv[i] = A[i*3+0]*trans[bn*3+0] + A[i*3+1]*trans[bn*3+1] + A[i*3+2]*trans[bn*3+2];
    float* o = mats + bn*21;
    for (int i = 0; i < 9; ++i) o[i]     = Pinv[i];
    for (int i = 0; i < 9; ++i) o[9+i]   = comb[i];
    for (int i = 0; i < 3; ++i) o[18+i]  = tv[i];
}

// ---------------------------------------------------------------------------
// Kernel 2: weight transpose + pad, K-pair interleaved:
//   wB2[((k>>1)*128 + n)*2 + (k&1)] = weight[n*512 + k]   (n<123, else 0)
// so one lane's (K, K+1) B-fragment is a single 8-byte coalesced load.
// ---------------------------------------------------------------------------
__global__ void setup_wb_kernel(const float* __restrict__ w, float* __restrict__ wB2) {
    int i = blockIdx.x * blockDim.x + threadIdx.x;   // i over CIN*OCPAD
    if (i >= CIN * OCPAD) return;
    int j  = i & 1;            // which K of the pair
    int n  = (i >> 1) & (OCPAD - 1);
    int kp = i >> 8;           // K pair index
    int k  = 2 * kp + j;
    wB2[i] = (n < OC) ? w[n * CIN + k] : 0.0f;
}

// ---------------------------------------------------------------------------
// Kernel 3: WMMA f32 GEMM (M=32 pixels, N=128, K=512) + fused bias + softmax.
// grid = (24, 22), block = 256 (8 waves). Wave w owns N-slice [16w, 16w+16)
// and computes TWO 16x16 M-tiles so each B fragment feeds two WMMAs.
// Fragment layouts (ISA 7.12.2, 32-bit 16x16x4):
//   A: lane L: M = L&15;  VGPR0: K = k0+2*(L>>4);  VGPR1: K = k0+2*(L>>4)+1
//   B: lane L: N = L&15;  same K striping as A
//   C/D: VGPR r, lane L -> M = r + 8*(L>>4), N = L&15
// LDS A is stored pair-interleaved [k/2][m][2] -> ds_load_b64 per fragment.
// ---------------------------------------------------------------------------
__global__ __launch_bounds__(256) void gemm_softmax_kernel(
        const float* __restrict__ x,       // [BN][CIN][FHW]
        const float* __restrict__ wB2,     // [CIN/2][OCPAD][2]
        const float* __restrict__ bias,    // [OC]
        float* __restrict__ depth_out,     // [BN][DEPTHC][FHW]
        float* __restrict__ feat_ws) {     // [BN][CT][FHW]
    __shared__ float lds_a[CIN * MTILE];       // 64 KB: A tile [k/2][m][2]
    __shared__ float lds_o[MTILE * OCPAD];     // 16 KB: out tile [m][n]

    const int bn     = blockIdx.x;
    const int m_base = blockIdx.y * MTILE;
    const int tid    = threadIdx.x;
    const float* xc  = x + (size_t)bn * CIN * FHW + m_base;

    // Stage A tile (512 K x 32 pixels), pair-interleaved for b64 reads.
    for (int i = tid; i < CIN * MTILE; i += 256) {
        int k = i >> 5, m = i & (MTILE - 1);
        lds_a[((k >> 1) * MTILE + m) * 2 + (k & 1)] = xc[k * FHW + m];
    }
    __syncthreads();

    const int lane   = tid & 31;
    const int lhi    = lane >> 4;          // 0/1: which K pair of the quad
    const int l15    = lane & 15;
    const int n_base = (tid >> 5) * 16;    // wave id * 16

    v8f acc0 = {};
    v8f acc1 = {};
    for (int k0 = 0; k0 < CIN; k0 += 4) {
        const int kp = (k0 >> 1) + lhi;    // K pair this half-wave consumes
        v2f a0 = *(const v2f*)&lds_a[(kp * MTILE + l15) * 2];
        v2f a1 = *(const v2f*)&lds_a[(kp * MTILE + 16 + l15) * 2];
        v2f b  = *(const v2f*)&wB2[((size_t)kp * OCPAD + n_base + l15) * 2];
        acc0 = __builtin_amdgcn_wmma_f32_16x16x4_f32(
                   false, a0, false, b, (short)0, acc0, false, false);
        acc1 = __builtin_amdgcn_wmma_f32_16x16x4_f32(
                   false, a1, false, b, (short)0, acc1, false, false);
    }

    // Spill both D tiles to LDS for the cross-wave softmax
#pragma unroll
    for (int r = 0; r < 8; ++r) {
        int m = r + 8 * lhi;
        lds_o[m * OCPAD + n_base + l15]        = acc0[r];
        lds_o[(16 + m) * OCPAD + n_base + l15] = acc1[r];
    }
    __syncthreads();

    // Softmax over the 59 depth logits (one thread per pixel; tiny)
    if (tid < MTILE) {
        float* row = &lds_o[tid * OCPAD];
        float mx = -3.4e38f;
        for (int o = 0; o < DEPTHC; ++o) {
            float v = row[o] + bias[o];
            row[o] = v;
            mx = fmaxf(mx, v);
        }
        float s = 0.0f;
        for (int o = 0; o < DEPTHC; ++o) {
            float e = __expf(row[o] - mx);
            row[o] = e;
            s += e;
        }
        float is = 1.0f / s;
        for (int o = 0; o < DEPTHC; ++o) row[o] *= is;
        for (int o = DEPTHC; o < OC; ++o) row[o] += bias[o];   // feat channels
    }
    __syncthreads();

    // depth -> d_out  (b,n,d,h,w)
    for (int i = tid; i < DEPTHC * MTILE; i += 256) {
        int o = i >> 5, m = i & (MTILE - 1);
        depth_out[((size_t)bn * DEPTHC + o) * FHW + m_base + m] = lds_o[m * OCPAD + o];
    }
    // feat -> workspace (bn,c,hw)
    for (int i = tid; i < CT * MTILE; i += 256) {
        int c = i >> 5, m = i & (MTILE - 1);
        feat_ws[((size_t)bn * CT + c) * FHW + m_base + m] = lds_o[m * OCPAD + DEPTHC + c];
    }
}

// ---------------------------------------------------------------------------
// Kernel 4: geometry + voxel pooling. One thread per (bn, d, pixel) point,
// 64 native f32 global atomics into the L2-resident BEV grid.
// ---------------------------------------------------------------------------
__global__ __launch_bounds__(256) void pool_kernel(
        const float* __restrict__ mats,
        const float* __restrict__ post_trans,   // [BN][3]
        const float* __restrict__ depth,        // [BN][DEPTHC][FHW]
        const float* __restrict__ feat,         // [BN][CT][FHW]
        float* __restrict__ bev) {              // [B][CT][NX0][NX1]
    const int NPTS = BN * DEPTHC * FHW;
    int pid = blockIdx.x * 256 + threadIdx.x;
    if (pid >= NPTS) return;

    int bn = pid / (DEPTHC * FHW);
    int r  = pid - bn * (DEPTHC * FHW);
    int d  = r / FHW;
    int m  = r - d * FHW;
    int h  = m / FW;
    int w  = m - h * FW;
    int b  = bn / NCAM;

    const float* M = mats + bn * 21;
    // frustum point
    float fx = (float)w * (703.0f / 43.0f);   // linspace(0, OGW-1, FW)
    float fy = (float)h * 17.0f;              // linspace(0, OGH-1, FH)
    float fz = 1.0f + (float)d;               // arange(1, 60, 1)
    // undo post augmentation
    float px = fx - post_trans[bn*3 + 0];
    float py = fy - post_trans[bn*3 + 1];
    float pz = fz - post_trans[bn*3 + 2];
    float q0 = M[0]*px + M[1]*py + M[2]*pz;
    float q1 = M[3]*px + M[4]*py + M[5]*pz;
    float q2 = M[6]*px + M[7]*py + M[8]*pz;
    // unproject: [u*z, v*z, z]
    float u0 = q0 * q2, u1 = q1 * q2, u2 = q2;
    // ego frame
    float g0 = M[9]*u0  + M[10]*u1 + M[11]*u2 + M[18];
    float g1 = M[12]*u0 + M[13]*u1 + M[14]*u2 + M[19];
    float g2 = M[15]*u0 + M[16]*u1 + M[17]*u2 + M[20];
    // voxelize: floor((g - (BX - DX/2)) / DX)
    int ix = (int)floorf((g0 + 51.2f) / 0.8f);
    int iy = (int)floorf((g1 + 51.2f) / 0.8f);
    int iz = (int)floorf((g2 + 10.0f) / 20.0f);
    if (ix < 0 || ix >= NX0 || iy < 0 || iy >= NX1 || iz != 0) return;

    float dv = depth[((size_t)bn * DEPTHC + d) * FHW + m];
    const float* f = feat + (size_t)bn * CT * FHW + m;
    float* out = bev + (((size_t)b * CT) * NX0 + ix) * NX1 + iy;
#pragma unroll 4
    for (int c = 0; c < CT; ++c) {
        unsafeAtomicAdd(out + (size_t)c * NX0 * NX1, dv * f[(size_t)c * FHW]);
    }
}

// ---------------------------------------------------------------------------
// Launch
// ---------------------------------------------------------------------------
extern "C" void kernel_launch(void* const* d_in, const int* in_sizes, int n_in,
                              void* d_out, int out_size, void* d_ws, size_t ws_size,
                              hipStream_t stream) {
    const float* x          = (const float*)d_in[0];
    const float* rots       = (const float*)d_in[1];
    const float* trans      = (const float*)d_in[2];
    const float* intrins    = (const float*)d_in[3];
    const float* post_rots  = (const float*)d_in[4];
    const float* post_trans = (const float*)d_in[5];
    const float* bda        = (const float*)d_in[6];
    const float* weight     = (const float*)d_in[7];
    const float* bias       = (const float*)d_in[8];

    float* out   = (float*)d_out;
    float* bev   = out;                          // 4*64*128*128
    float* depth = out + BEV_ELEMS;              // 24*59*704

    float* ws   = (float*)d_ws;
    float* mats = ws;                            // 24*21, padded to 1024
    float* wB2  = ws + 1024;                     // 512*128 (pair-interleaved)
    float* feat = ws + 1024 + CIN * OCPAD;       // 24*64*704

    hipMemsetAsync(bev, 0, BEV_ELEMS * sizeof(float), stream);

    setup_mats_kernel<<<1, 32, 0, stream>>>(rots, trans, intrins, post_rots, bda, mats);
    setup_wb_kernel<<<(CIN * OCPAD + 255) / 256, 256, 0, stream>>>(weight, wB2);

    dim3 g(BN, FHW / MTILE);
    gemm_softmax_kernel<<<g, 256, 0, stream>>>(x, wB2, bias, depth, feat);

    int npts = BN * DEPTHC * FHW;
    pool_kernel<<<(npts + 255) / 256, 256, 0, stream>>>(mats, post_trans, depth, feat, bev);
}